// YOLOLoss_84670985273983
// MI455X (gfx1250) — compile-verified
//
#include <hip/hip_runtime.h>
#include <cstdint>

// ---------------------------------------------------------------------------
// YOLO loss on MI455X (gfx1250).
// Memory-bound: 166 MB read / 23.3 TB/s ~= 7.1 us floor.
// - Global->LDS async b128 copies (gfx1250 async path, ASYNCcnt) give fully
//   coalesced HBM traffic; the 60B-strided per-cell gathers hit LDS instead.
// - v_wmma_f32_16x16x4_f32 against a ones matrix performs the 32-lane
//   reduction in f32 (D[m][n] = val[m] + val[m+16]; 8 adds + shfl_xor(16)).
// ---------------------------------------------------------------------------

#define PSZ 15                      // 5 + NUM_CLASSES
#define THREADS 256
#define CELLS_PER_TILE 512
#define FLOATS_PER_ARR (CELLS_PER_TILE * PSZ)          // 7680 floats = 30720 B
#define B128_PER_ARR   (FLOATS_PER_ARR / 4)            // 1920 x 16B chunks
#define TOTAL_CELLS    (2048 * 26 * 26)                // 1,384,448
#define NUM_TILES      (TOTAL_CELLS / CELLS_PER_TILE)  // 2704 exactly

typedef __attribute__((ext_vector_type(2))) float v2f;
typedef __attribute__((ext_vector_type(8))) float v8f;

// LDS byte offset of a shared-memory pointer (addrspacecast strips aperture).
__device__ __forceinline__ uint32_t lds_off_u32(const void* p) {
  return (uint32_t)(uintptr_t)(__attribute__((address_space(3))) const void*)p;
}

// One async 16B global->LDS copy. GVS mode: 64-bit SGPR base + 32-bit VGPR off.
__device__ __forceinline__ void async_ld_b128(uint32_t lds_addr, uint32_t voff,
                                              const float* base) {
  asm volatile("global_load_async_to_lds_b128 %0, %1, %2"
               :: "v"(lds_addr), "v"(voff), "s"(base)
               : "memory");
}

// Full-wave (32 lane) sum via WMMA f32 16x16x4 with B = ones.
// A[m][0]=val[m] (lanes 0-15, VGPR0), A[m][2]=val[m+16] (lanes 16-31, VGPR0);
// VGPR1 slots are zero. D[m][n] = val[m] + val[m+16].
// Lane n<16 holds D[0..7][n] in c[0..7]; lane n>=16 holds D[8..15][n-16].
// Sum 8 VGPRs, then xor-shuffle with lane^16 -> every lane has the total.
__device__ __forceinline__ float wave_sum32(float v) {
  v2f a; a.x = v;    a.y = 0.0f;
  v2f b; b.x = 1.0f; b.y = 1.0f;
  v8f c = {0.f, 0.f, 0.f, 0.f, 0.f, 0.f, 0.f, 0.f};
  c = __builtin_amdgcn_wmma_f32_16x16x4_f32(
      /*neg_a=*/false, a, /*neg_b=*/false, b,
      /*c_mod=*/(short)0, c, /*reuse_a=*/false, /*reuse_b=*/false);
  float s = ((c[0] + c[1]) + (c[2] + c[3])) + ((c[4] + c[5]) + (c[6] + c[7]));
  s += __shfl_xor(s, 16, 32);
  return s;
}

__global__ __launch_bounds__(THREADS)
void yolo_partial_kernel(const float* __restrict__ pred,
                         const float* __restrict__ tgt,
                         float* __restrict__ ws) {
  __shared__ float ldsP[FLOATS_PER_ARR];
  __shared__ float ldsT[FLOATS_PER_ARR];
  __shared__ float wredA[8], wredB[8], wredC[8];

  const int tid = threadIdx.x;
  const int tile = blockIdx.x;
  const uint32_t tileByte = (uint32_t)tile * (uint32_t)(FLOATS_PER_ARR * 4);

  const uint32_t lpBase = lds_off_u32(ldsP);
  const uint32_t ltBase = lds_off_u32(ldsT);

  // Stage both arrays' tile into LDS: contiguous b128 async copies.
  for (int i = tid; i < B128_PER_ARR; i += THREADS) {
    const uint32_t o = (uint32_t)i * 16u;
    async_ld_b128(lpBase + o, tileByte + o, pred);
    async_ld_b128(ltBase + o, tileByte + o, tgt);
  }
  asm volatile("s_wait_asynccnt 0" ::: "memory");
  __syncthreads();

  // Per-cell loss terms; raw sums (lambda / class-mean scaling applied at end
  // to match the reference's  lambda * sum(...)  summation order).
  float coordS = 0.f, confS = 0.f, clsS = 0.f;
#pragma unroll
  for (int k = 0; k < CELLS_PER_TILE / THREADS; ++k) {
    const int c = tid * (CELLS_PER_TILE / THREADS) + k;
    const float* p = &ldsP[c * PSZ];
    const float* t = &ldsT[c * PSZ];
    const bool obj = t[4] > 0.0f;

    float cs = 0.f;
#pragma unroll
    for (int j = 0; j < 4; ++j) { const float d = p[j] - t[j]; cs += d * d; }

    const float sp = 1.0f / (1.0f + __expf(-p[4]));   // sigmoid(conf)
    const float dc = sp - t[4];

    float ks = 0.f;
#pragma unroll
    for (int j = 5; j < 15; ++j) { const float d = p[j] - t[j]; ks += d * d; }

    if (obj) { coordS += cs; confS += dc * dc; clsS += ks; }
    else     { confS += 0.5f * sp * sp; }               // LAMBDA_NOOBJ = 0.5
  }

  // Wave reduce (WMMA), then cross-wave via LDS.
  coordS = wave_sum32(coordS);
  confS  = wave_sum32(confS);
  clsS   = wave_sum32(clsS);

  const int lane = tid & 31, wid = tid >> 5;
  if (lane == 0) { wredA[wid] = coordS; wredB[wid] = confS; wredC[wid] = clsS; }
  __syncthreads();
  if (tid == 0) {
    float a = 0.f, b = 0.f, d = 0.f;
#pragma unroll
    for (int w = 0; w < 8; ++w) { a += wredA[w]; b += wredB[w]; d += wredC[w]; }
    ((float4*)ws)[tile] = make_float4(a, b, d, 0.f);
  }
}

__global__ __launch_bounds__(THREADS)
void yolo_final_kernel(const float* __restrict__ ws, float* __restrict__ out) {
  __shared__ float wredA[8], wredB[8], wredC[8];
  const int tid = threadIdx.x;

  float a = 0.f, b = 0.f, d = 0.f;
  for (int i = tid; i < NUM_TILES; i += THREADS) {
    const float4 v = ((const float4*)ws)[i];
    a += v.x; b += v.y; d += v.z;
  }

  a = wave_sum32(a);
  b = wave_sum32(b);
  d = wave_sum32(d);

  const int lane = tid & 31, wid = tid >> 5;
  if (lane == 0) { wredA[wid] = a; wredB[wid] = b; wredC[wid] = d; }
  __syncthreads();
  if (tid == 0) {
    float A = 0.f, B = 0.f, D = 0.f;
#pragma unroll
    for (int w = 0; w < 8; ++w) { A += wredA[w]; B += wredB[w]; D += wredC[w]; }
    const float coord = 5.0f * A;   // LAMBDA_COORD
    const float conf  = B;
    const float cls   = 0.1f * D;   // mean over NUM_CLASSES = 10
    out[0] = coord + conf + cls;
    out[1] = coord;
    out[2] = conf;
    out[3] = cls;
  }
}

extern "C" void kernel_launch(void* const* d_in, const int* in_sizes, int n_in,
                              void* d_out, int out_size, void* d_ws, size_t ws_size,
                              hipStream_t stream) {
  const float* pred = (const float*)d_in[0];   // (B, S, S, 15) f32
  const float* tgt  = (const float*)d_in[1];   // (B, S*S*15)   f32, same flat layout
  float* ws = (float*)d_ws;                    // NUM_TILES * float4 partials

  yolo_partial_kernel<<<NUM_TILES, THREADS, 0, stream>>>(pred, tgt, ws);
  yolo_final_kernel<<<1, THREADS, 0, stream>>>(ws, (float*)d_out);
}